// FeaturePropagation_28355374088836
// MI455X (gfx1250) — compile-verified
//
#include <hip/hip_runtime.h>
#include <hip/hip_bf16.h>

typedef __attribute__((ext_vector_type(2))) float v2f;
typedef __attribute__((ext_vector_type(8))) float v8f;

#define B_   8
#define C1_  256
#define M_   4096
#define C2_  128
#define N_   16384
#define CO_  256
#define K_   (C1_ + C2_)   /* 384 */

// -----------------------------------------------------------------------------
// Kernel 1: fused gather + concat + GEMM (V_WMMA_F32_16X16X4_F32) + bias.
// Grid: (N/128, CO/64, B). Block: 256 threads = 8 waves.
// Each wave owns a 16-wide n-slice and 4 o-tiles (64 output channels),
// reusing each B fragment across 4 WMMAs.
// -----------------------------------------------------------------------------
__global__ __launch_bounds__(256) void fp_gemm_kernel(
    const float* __restrict__ feature,   // [B, C1, M]
    const float* __restrict__ enc,       // [B, C2, N]
    const int*   __restrict__ idx,       // [B, N] (int32)
    const float* __restrict__ W,         // [K, CO]
    const float* __restrict__ bias,      // [CO]
    float* __restrict__ y)               // [B, CO, N] (pre-BN)
{
    const int tid  = threadIdx.x;
    const int wave = tid >> 5;
    const int lane = tid & 31;
    const int col  = lane & 15;   // row index (A) and n index (B/C) within tile
    const int khi  = lane >> 4;   // which K pair this half-wave holds

    const int bb     = blockIdx.z;
    const int o_base = blockIdx.y * 64;
    const int n      = blockIdx.x * 128 + wave * 16 + col;

    const long gidx = (long)idx[bb * N_ + n];
    const float* __restrict__ featb = feature + (size_t)bb * C1_ * M_;
    const float* __restrict__ encb  = enc     + (size_t)bb * C2_ * N_;

    v8f acc0 = {}; v8f acc1 = {}; v8f acc2 = {}; v8f acc3 = {};

    // ---- K part 1: encoder channels k in [0,128), streamed (coalesced in n)
    for (int k = 0; k < C2_; k += 4) {
        const int kk = k + 2 * khi;           // this half-wave's K pair
        v2f bfrag;
        bfrag.x = encb[(size_t)kk * N_ + n];
        bfrag.y = encb[(size_t)(kk + 1) * N_ + n];

        const float* w0 = W + (size_t)kk       * CO_ + o_base + col;
        const float* w1 = W + (size_t)(kk + 1) * CO_ + o_base + col;
        v2f a0; a0.x = w0[0];  a0.y = w1[0];
        v2f a1; a1.x = w0[16]; a1.y = w1[16];
        v2f a2; a2.x = w0[32]; a2.y = w1[32];
        v2f a3; a3.x = w0[48]; a3.y = w1[48];

        acc0 = __builtin_amdgcn_wmma_f32_16x16x4_f32(false, a0, false, bfrag, (short)0, acc0, false, false);
        acc1 = __builtin_amdgcn_wmma_f32_16x16x4_f32(false, a1, false, bfrag, (short)0, acc1, false, false);
        acc2 = __builtin_amdgcn_wmma_f32_16x16x4_f32(false, a2, false, bfrag, (short)0, acc2, false, false);
        acc3 = __builtin_amdgcn_wmma_f32_16x16x4_f32(false, a3, false, bfrag, (short)0, acc3, false, false);
    }

    // ---- K part 2: gathered feature channels k in [128,384)
    for (int k = C2_; k < K_; k += 4) {
        const int kk = k + 2 * khi;
        const int c  = kk - C2_;
        v2f bfrag;
        bfrag.x = featb[(size_t)c       * M_ + gidx];
        bfrag.y = featb[(size_t)(c + 1) * M_ + gidx];

        const float* w0 = W + (size_t)kk       * CO_ + o_base + col;
        const float* w1 = W + (size_t)(kk + 1) * CO_ + o_base + col;
        v2f a0; a0.x = w0[0];  a0.y = w1[0];
        v2f a1; a1.x = w0[16]; a1.y = w1[16];
        v2f a2; a2.x = w0[32]; a2.y = w1[32];
        v2f a3; a3.x = w0[48]; a3.y = w1[48];

        acc0 = __builtin_amdgcn_wmma_f32_16x16x4_f32(false, a0, false, bfrag, (short)0, acc0, false, false);
        acc1 = __builtin_amdgcn_wmma_f32_16x16x4_f32(false, a1, false, bfrag, (short)0, acc1, false, false);
        acc2 = __builtin_amdgcn_wmma_f32_16x16x4_f32(false, a2, false, bfrag, (short)0, acc2, false, false);
        acc3 = __builtin_amdgcn_wmma_f32_16x16x4_f32(false, a3, false, bfrag, (short)0, acc3, false, false);
    }

    // ---- Epilogue: C/D layout -> lane (0-15: M=r, 16-31: M=r+8), N = col
    const size_t out_b = (size_t)bb * CO_;
    #pragma unroll
    for (int r = 0; r < 8; ++r) {
        const int o0 = o_base + r + khi * 8;
        y[(out_b + (o0 +  0)) * N_ + n] = acc0[r] + bias[o0 +  0];
        y[(out_b + (o0 + 16)) * N_ + n] = acc1[r] + bias[o0 + 16];
        y[(out_b + (o0 + 32)) * N_ + n] = acc2[r] + bias[o0 + 32];
        y[(out_b + (o0 + 48)) * N_ + n] = acc3[r] + bias[o0 + 48];
    }
}

// -----------------------------------------------------------------------------
// Kernel 2: per-channel batch stats over (B,N); folds gamma/beta into
// scale/shift pairs written to workspace. One block per channel.
// -----------------------------------------------------------------------------
__global__ __launch_bounds__(256) void fp_stats_kernel(
    const float* __restrict__ y,      // [B, CO, N]
    float* __restrict__ ss,           // [CO][2] -> scale, shift
    const float* __restrict__ gamma,
    const float* __restrict__ beta)
{
    const int o   = blockIdx.x;
    const int tid = threadIdx.x;
    float s = 0.f, s2 = 0.f;
    for (int b = 0; b < B_; ++b) {
        const float* p = y + ((size_t)b * CO_ + o) * N_;
        for (int nn = tid * 4; nn < N_; nn += 256 * 4) {
            float4 v = *(const float4*)(p + nn);
            s  += v.x + v.y + v.z + v.w;
            s2 += v.x * v.x + v.y * v.y + v.z * v.z + v.w * v.w;
        }
    }
    __shared__ float red0[256];
    __shared__ float red1[256];
    red0[tid] = s; red1[tid] = s2;
    __syncthreads();
    for (int off = 128; off > 0; off >>= 1) {
        if (tid < off) {
            red0[tid] += red0[tid + off];
            red1[tid] += red1[tid + off];
        }
        __syncthreads();
    }
    if (tid == 0) {
        const float inv  = 1.0f / (float)((size_t)B_ * N_);
        const float mean = red0[0] * inv;
        const float var  = red1[0] * inv - mean * mean;
        const float rstd = rsqrtf(var + 1e-5f);
        const float scale = gamma[o] * rstd;
        ss[2 * o]     = scale;
        ss[2 * o + 1] = beta[o] - mean * scale;
    }
}

// -----------------------------------------------------------------------------
// Kernel 3: in-place normalize + LeakyReLU(0.2), float4 vectorized.
// Channel is uniform within each float4 (N is a multiple of 4).
// -----------------------------------------------------------------------------
__global__ __launch_bounds__(256) void fp_bn_lrelu_kernel(
    float* __restrict__ y, const float* __restrict__ ss)
{
    const size_t total4 = (size_t)B_ * CO_ * N_ / 4;
    for (size_t i = (size_t)blockIdx.x * 256 + threadIdx.x; i < total4;
         i += (size_t)gridDim.x * 256) {
        const size_t e = i * 4;
        const int o = (int)((e >> 14) & (CO_ - 1));  // (e / N) % CO
        const float scale = ss[2 * o];
        const float shift = ss[2 * o + 1];
        float4 v = ((const float4*)y)[i];
        v.x = v.x * scale + shift;
        v.y = v.y * scale + shift;
        v.z = v.z * scale + shift;
        v.w = v.w * scale + shift;
        v.x = v.x > 0.f ? v.x : 0.2f * v.x;
        v.y = v.y > 0.f ? v.y : 0.2f * v.y;
        v.z = v.z > 0.f ? v.z : 0.2f * v.z;
        v.w = v.w > 0.f ? v.w : 0.2f * v.w;
        ((float4*)y)[i] = v;
    }
}

extern "C" void kernel_launch(void* const* d_in, const int* in_sizes, int n_in,
                              void* d_out, int out_size, void* d_ws, size_t ws_size,
                              hipStream_t stream) {
    const float* feature = (const float*)d_in[0];  // [8,256,4096]
    const float* enc     = (const float*)d_in[1];  // [8,128,16384]
    const int*   idx     = (const int*)  d_in[2];  // [8,16384] int32
    const float* W       = (const float*)d_in[3];  // [384,256]
    const float* bias    = (const float*)d_in[4];  // [256]
    const float* gamma   = (const float*)d_in[5];  // [256]
    const float* beta    = (const float*)d_in[6];  // [256]
    float* y  = (float*)d_out;                     // [8,256,16384]
    float* ss = (float*)d_ws;                      // 256*2 floats scratch

    dim3 g1(N_ / 128, CO_ / 64, B_);
    fp_gemm_kernel<<<g1, 256, 0, stream>>>(feature, enc, idx, W, bias, y);
    fp_stats_kernel<<<CO_, 256, 0, stream>>>(y, ss, gamma, beta);
    fp_bn_lrelu_kernel<<<4096, 256, 0, stream>>>(y, ss);
}